// FNOGeoEncoder_33483565039961
// MI455X (gfx1250) — compile-verified
//
#include <hip/hip_runtime.h>
#include <hip/hip_bf16.h>
#include <math.h>

// FNO geo encoder, exact mode-0 reduction (see analysis):
//   S[k]   = sum_n V[n,k]
//   x0[c]  = fc0_w[c,:].S + N*fc0_b[c]
//   x_{i+1}[o] = sum_c x_i[c] * sc_w[i,c,o,0,RE]      (4 layers, f32 WMMA)
//   out    = tanh(x4/N @ fc1^T + b1) @ fc2^T + b2

#define NPTS 131072
#define WIDTH 256
#define MODES 512
// floats per layer of sc_w: 256*256*512*2
#define LAYER_STRIDE 67108864ull
// float stride between c (input channel) for fixed o: 256*512*2
#define C_STRIDE 262144ull
// float stride between o (output channel): 512*2
#define O_STRIDE 1024ull

typedef __attribute__((ext_vector_type(2))) float v2f;
typedef __attribute__((ext_vector_type(8))) float v8f;

// ---------------- Phase 1: partial column sums of V (256 blocks) -------------
__global__ void fno_reduce_v(const float* __restrict__ V, float* __restrict__ part) {
    __shared__ float s[256][3];
    const int t = threadIdx.x;
    const int rows = NPTS / 256;                 // 512 rows per block
    const size_t base = (size_t)blockIdx.x * rows;
    float a0 = 0.f, a1 = 0.f, a2 = 0.f;
    for (int r = t; r < rows; r += 256) {
        size_t idx = (base + r) * 3u;
        a0 += V[idx + 0];
        a1 += V[idx + 1];
        a2 += V[idx + 2];
    }
    s[t][0] = a0; s[t][1] = a1; s[t][2] = a2;
    __syncthreads();
    for (int stride = 128; stride > 0; stride >>= 1) {   // fixed-order tree
        if (t < stride) {
            s[t][0] += s[t + stride][0];
            s[t][1] += s[t + stride][1];
            s[t][2] += s[t + stride][2];
        }
        __syncthreads();
    }
    if (t == 0) {
        part[blockIdx.x * 3 + 0] = s[0][0];
        part[blockIdx.x * 3 + 1] = s[0][1];
        part[blockIdx.x * 3 + 2] = s[0][2];
    }
}

// --------- Phase 2: finish S, lift to x0[c] = fc0_w[c,:].S + N*b[c] ----------
__global__ void fno_lift(const float* __restrict__ part,
                         const float* __restrict__ fc0_w,
                         const float* __restrict__ fc0_b,
                         float* __restrict__ x0) {
    __shared__ float s[256][3];
    __shared__ float S[3];
    const int t = threadIdx.x;                   // 256 threads
    s[t][0] = part[t * 3 + 0];
    s[t][1] = part[t * 3 + 1];
    s[t][2] = part[t * 3 + 2];
    __syncthreads();
    for (int stride = 128; stride > 0; stride >>= 1) {
        if (t < stride) {
            s[t][0] += s[t + stride][0];
            s[t][1] += s[t + stride][1];
            s[t][2] += s[t + stride][2];
        }
        __syncthreads();
    }
    if (t == 0) { S[0] = s[0][0]; S[1] = s[0][1]; S[2] = s[0][2]; }
    __syncthreads();
    x0[t] = fc0_w[t * 3 + 0] * S[0] + fc0_w[t * 3 + 1] * S[1] +
            fc0_w[t * 3 + 2] * S[2] + (float)NPTS * fc0_b[t];
}

// --------- Spectral layer (mode 0, real part): y[o] = sum_c x[c]*W[c,o] ------
// Grid: 16 blocks (one 16-wide o-tile each), 128 threads = 4 waves.
// D = A x B accumulated over K: A(16x4 f32) = W^T tile (M=o, K=c),
// B(4x16) = x broadcast over N. Wave w covers kb = w*16 .. w*16+15 (c = 4*kb..).
__global__ void fno_layer(const float* __restrict__ W,   // layer base of sc_w
                          const float* __restrict__ xin,
                          float* __restrict__ xout) {
    __shared__ float xs[WIDTH];
    __shared__ float partial[4][32][8];
    const int t = threadIdx.x;
    xs[t]       = xin[t];
    xs[t + 128] = xin[t + 128];
    __syncthreads();

    const int wave = t >> 5;
    const int lane = t & 31;
    const int m    = lane & 15;            // M index (o within tile)
    const int hi   = (lane >> 4) & 1;      // lane half selects K pair (0,1 or 2,3)
    const int o    = blockIdx.x * 16 + m;
    const float* Wo = W + (size_t)o * O_STRIDE;   // + c*C_STRIDE reaches W[c,o]

    v8f acc = {0.f, 0.f, 0.f, 0.f, 0.f, 0.f, 0.f, 0.f};
    const int kb0 = wave * 16;
    #pragma unroll 4
    for (int kb = kb0; kb < kb0 + 16; ++kb) {
        // A layout (ISA 16x4 f32): VGPR0 = K{0|2}, VGPR1 = K{1|3}; M = lane&15
        const int c0 = 4 * kb + 2 * hi;    // K = 0 or 2
        const int c1 = c0 + 1;             // K = 1 or 3
        v2f a, b;
        a[0] = Wo[(size_t)c0 * C_STRIDE];
        a[1] = Wo[(size_t)c1 * C_STRIDE];
        b[0] = xs[c0];                     // B = x broadcast across all N cols
        b[1] = xs[c1];
        acc = __builtin_amdgcn_wmma_f32_16x16x4_f32(
            /*neg_a=*/false, a, /*neg_b=*/false, b,
            /*c_mod=*/(short)0, acc, /*reuse_a=*/false, /*reuse_b=*/false);
    }
    #pragma unroll
    for (int r = 0; r < 8; ++r) partial[wave][lane][r] = acc[r];
    __syncthreads();

    // Fixed-order cross-wave combine; D layout: lanes 0-15 VGPR r -> M=r,
    // lanes 16-31 VGPR r -> M=8+r (N-replicated, so lane 0 / lane 16 suffice).
    if (t == 0) {
        for (int r = 0; r < 8; ++r) {
            float s0 = ((partial[0][0][r]  + partial[1][0][r]) +
                        (partial[2][0][r]  + partial[3][0][r]));
            float s1 = ((partial[0][16][r] + partial[1][16][r]) +
                        (partial[2][16][r] + partial[3][16][r]));
            xout[blockIdx.x * 16 + r]     = s0;
            xout[blockIdx.x * 16 + 8 + r] = s1;
        }
    }
}

// ------------------------- Head MLP (single block) ---------------------------
__global__ void fno_head(const float* __restrict__ x4,
                         const float* __restrict__ fc1_w,
                         const float* __restrict__ fc1_b,
                         const float* __restrict__ fc2_w,
                         const float* __restrict__ fc2_b,
                         float* __restrict__ out) {
    __shared__ float feat[WIDTH];
    __shared__ float h[128];
    const int t = threadIdx.x;                 // 256 threads
    feat[t] = x4[t] * (1.0f / (float)NPTS);
    __syncthreads();
    if (t < 128) {
        float acc = fc1_b[t];
        for (int c = 0; c < WIDTH; ++c) acc += feat[c] * fc1_w[t * WIDTH + c];
        h[t] = tanhf(acc);
    }
    __syncthreads();
    float acc = fc2_b[t];
    for (int j = 0; j < 128; ++j) acc += h[j] * fc2_w[t * 128 + j];
    out[t] = acc;
}

extern "C" void kernel_launch(void* const* d_in, const int* in_sizes, int n_in,
                              void* d_out, int out_size, void* d_ws, size_t ws_size,
                              hipStream_t stream) {
    const float* V     = (const float*)d_in[0];
    const float* fc0_w = (const float*)d_in[1];
    const float* fc0_b = (const float*)d_in[2];
    const float* sc_w  = (const float*)d_in[3];
    const float* fc1_w = (const float*)d_in[4];
    const float* fc1_b = (const float*)d_in[5];
    const float* fc2_w = (const float*)d_in[6];
    const float* fc2_b = (const float*)d_in[7];

    float* ws   = (float*)d_ws;
    float* part = ws;              // 768 floats
    float* xa   = ws + 1024;       // 256 floats
    float* xb   = ws + 1536;       // 256 floats

    fno_reduce_v<<<256, 256, 0, stream>>>(V, part);
    fno_lift<<<1, 256, 0, stream>>>(part, fc0_w, fc0_b, xa);

    const float* cur = xa;
    float* nxt = xb;
    for (int i = 0; i < 4; ++i) {
        fno_layer<<<16, 128, 0, stream>>>(sc_w + (size_t)i * LAYER_STRIDE, cur, nxt);
        float* tmp = (float*)cur;
        cur = nxt;
        nxt = tmp;
    }
    // after 4 swaps cur == xa
    fno_head<<<1, 256, 0, stream>>>(cur, fc1_w, fc1_b, fc2_w, fc2_b, (float*)d_out);
}